// TensorGNAN_44796508897983
// MI455X (gfx1250) — compile-verified
//
#include <hip/hip_runtime.h>

typedef __attribute__((ext_vector_type(2))) float v2f;
typedef __attribute__((ext_vector_type(8))) float v8f;

#define Nn 4096
#define Ff 64
#define Hh 16
#define Cc 7
#define KSPLIT 16
#define TILES (Nn / 16)        // 256 row tiles
#define KCHUNK (Nn / KSPLIT)   // 256 columns per wave

// ---------------------------------------------------------------------------
// Stage 1: g[n,c] = sum_f sum_h relu(x*w1+b1)*w2 + sum_f b2   -> [4096 x 7]
// Stored in "pair" layout so stage 2 can fetch each lane's two consecutive
// K-rows of the WMMA B tile with a single 8-byte load:
//   g[n][c] -> gws[(n>>1)*32 + c*2 + (n&1)], columns 7..15 zeroed.
// ---------------------------------------------------------------------------
__global__ void gnan_g_kernel(const float* __restrict__ x,
                              const float* __restrict__ f_w1,
                              const float* __restrict__ f_b1,
                              const float* __restrict__ f_w2,
                              const float* __restrict__ f_b2,
                              float* __restrict__ gws) {
    int n = blockIdx.x * blockDim.x + threadIdx.x;
    if (n >= Nn) return;
    float acc[Cc];
#pragma unroll
    for (int c = 0; c < Cc; ++c) acc[c] = 0.f;
    for (int f = 0; f < Ff; ++f) {
        float xv = x[n * Ff + f];
#pragma unroll
        for (int h = 0; h < Hh; ++h) {
            float hv = fmaxf(xv * f_w1[f * Hh + h] + f_b1[f * Hh + h], 0.f);
#pragma unroll
            for (int c = 0; c < Cc; ++c)
                acc[c] += hv * f_w2[(f * Hh + h) * Cc + c];
        }
#pragma unroll
        for (int c = 0; c < Cc; ++c) acc[c] += f_b2[f * Cc + c];
    }
    int base = (n >> 1) * 32 + (n & 1);
#pragma unroll
    for (int c = 0; c < 16; ++c)
        gws[base + c * 2] = (c < Cc) ? acc[c] : 0.f;
}

// ---------------------------------------------------------------------------
// Stage 2: out_tile += M_tile(16x4) @ g_tile(4x16) via V_WMMA_F32_16X16X4_F32,
// with M = rho(dist/norm) materialized on the fly in registers.
// A layout (ISA 32-bit A 16x4): lanes 0-15 hold {K0,K1} of row M=lane,
//                               lanes 16-31 hold {K2,K3} of row M=lane-16.
// B layout: lanes 0-15 hold rows {K0,K1} at col N=lane; lanes 16-31 {K2,K3}.
// 256 tiles x 16 K-splits = 4096 waves; K loop unrolled x2 so each body
// issues one clause of 6 global_load_b64 + ~200 dual-issue VALU + 2 WMMA.
// ---------------------------------------------------------------------------
__global__ void gnan_mm_kernel(const float* __restrict__ dist,
                               const float* __restrict__ norm,
                               const float* __restrict__ rho_w1,
                               const float* __restrict__ rho_b1,
                               const float* __restrict__ rho_w2,
                               const float* __restrict__ rho_b2,
                               const float* __restrict__ gws,
                               float* __restrict__ part) {
    const int lane  = threadIdx.x & 31;
    const int gwave = blockIdx.x * (blockDim.x >> 5) + (threadIdx.x >> 5);
    const int tile  = gwave / KSPLIT;   // 0..255
    const int split = gwave % KSPLIT;   // 0..15
    const int n0      = tile * 16;
    const int k_begin = split * KCHUNK;

    // rho weights are wave-uniform -> scalar loads
    float w1[Hh], b1[Hh], w2[Hh];
#pragma unroll
    for (int h = 0; h < Hh; ++h) {
        w1[h] = rho_w1[h];
        b1[h] = rho_b1[h];
        w2[h] = rho_w2[h];
    }
    const float b2 = rho_b2[0];

    const int r    = lane & 15;          // row within tile
    const int koff = (lane >> 4) << 1;   // 0 (lanes 0-15) or 2 (lanes 16-31)
    const float* drow = dist + (size_t)(n0 + r) * Nn;
    const float* nrow = norm + (size_t)(n0 + r) * Nn;

    v8f acc = {};
#pragma unroll 2
    for (int k0 = k_begin; k0 < k_begin + KCHUNK; k0 += 4) {
        const int kk = k0 + koff;
        v2f dv = *(const v2f*)(drow + kk);
        v2f nv = *(const v2f*)(nrow + kk);
        v2f b  = *(const v2f*)(gws + (size_t)(kk >> 1) * 32 + (lane & 15) * 2);
        float t0 = dv.x * __builtin_amdgcn_rcpf(nv.x);
        float t1 = dv.y * __builtin_amdgcn_rcpf(nv.y);
        float m0 = b2, m1 = b2;
#pragma unroll
        for (int h = 0; h < Hh; ++h) {
            m0 += fmaxf(t0 * w1[h] + b1[h], 0.f) * w2[h];
            m1 += fmaxf(t1 * w1[h] + b1[h], 0.f) * w2[h];
        }
        v2f a;
        a.x = m0;
        a.y = m1;
        // (neg_a, A, neg_b, B, c_mod, C, reuse_a, reuse_b)
        acc = __builtin_amdgcn_wmma_f32_16x16x4_f32(
            false, a, false, b, (short)0, acc, false, false);
    }

    // partial tile: part[(tile*KSPLIT+split)][vgpr][lane]
    float* p = part + (size_t)(tile * KSPLIT + split) * 256;
#pragma unroll
    for (int v = 0; v < 8; ++v)
        p[v * 32 + lane] = acc[v];
}

// ---------------------------------------------------------------------------
// Stage 3: deterministic reduction over the 16 K-splits; de-swizzle the WMMA
// C/D layout (lanes 0-15: M=vgpr, N=lane; lanes 16-31: M=8+vgpr, N=lane-16)
// and write only the 7 valid classes.
// ---------------------------------------------------------------------------
__global__ void gnan_reduce_kernel(const float* __restrict__ part,
                                   float* __restrict__ out) {
    int idx = blockIdx.x * blockDim.x + threadIdx.x;
    if (idx >= Nn * Cc) return;
    int n = idx / Cc, c = idx % Cc;
    int tile = n >> 4;
    int row  = n & 15;
    int lane = c + ((row >> 3) << 4);
    int v    = row & 7;
    const float* p = part + (size_t)tile * KSPLIT * 256;
    float s = 0.f;
#pragma unroll
    for (int sp = 0; sp < KSPLIT; ++sp)
        s += p[sp * 256 + v * 32 + lane];
    out[idx] = s;
}

extern "C" void kernel_launch(void* const* d_in, const int* in_sizes, int n_in,
                              void* d_out, int out_size, void* d_ws, size_t ws_size,
                              hipStream_t stream) {
    const float* x      = (const float*)d_in[0];
    // d_in[1] = edge_index (int64) -- unused by the math
    const float* dist   = (const float*)d_in[2];
    const float* norm   = (const float*)d_in[3];
    const float* f_w1   = (const float*)d_in[4];
    const float* f_b1   = (const float*)d_in[5];
    const float* f_w2   = (const float*)d_in[6];
    const float* f_b2   = (const float*)d_in[7];
    const float* rho_w1 = (const float*)d_in[8];
    const float* rho_b1 = (const float*)d_in[9];
    const float* rho_w2 = (const float*)d_in[10];
    const float* rho_b2 = (const float*)d_in[11];
    float* out = (float*)d_out;

    float* gws  = (float*)d_ws;          // 4096*16 floats = 256 KB
    float* part = gws + Nn * 16;         // 256*16*256 floats = 4 MB

    gnan_g_kernel<<<Nn / 256, 256, 0, stream>>>(x, f_w1, f_b1, f_w2, f_b2, gws);

    // 256 tiles * 16 splits = 4096 waves; 4 waves (128 threads) per block
    gnan_mm_kernel<<<(TILES * KSPLIT) / 4, 128, 0, stream>>>(
        dist, norm, rho_w1, rho_b1, rho_w2, rho_b2, gws, part);

    gnan_reduce_kernel<<<(Nn * Cc + 255) / 256, 256, 0, stream>>>(part, out);
}